// PaperLightGCN_87668872446394
// MI455X (gfx1250) — compile-verified
//
#include <hip/hip_runtime.h>
#include <hip/hip_bf16.h>
#include <stdint.h>

// ---------------------------------------------------------------------------
// LightGCN propagation on MI455X (gfx1250).
//   out = (ego + A*ego + A^2*ego + A^3*ego) / 4,  A given as COO (4.8M nnz).
// Memory-bound SpMM: both embedding buffers (38.4 MB each) are L2-resident
// (192 MB L2); edge stream (57.6 MB/layer) is HBM-streamed via async-to-LDS
// NT loads; scatter uses native f32 global atomics (no return -> STOREcnt).
// Final average uses chained v_wmma_f32_16x16x4_f32 (free at this bandwidth).
// ---------------------------------------------------------------------------

#define USER_NUM 100000
#define ITEM_NUM 50000
#define N_NODES  (USER_NUM + ITEM_NUM)     // 150000
#define EMB_DIM  64
#define NNZ      4800000
#define NELEM    (N_NODES * EMB_DIM)       // 9,600,000 floats = 38.4 MB

#define SPMM_BLOCK 256
#define WPB        (SPMM_BLOCK / 32)       // waves per block
#define BATCH      32                      // edges per wave per stage

static_assert((NNZ % BATCH) == 0, "edge batches are exact");
static_assert((N_NODES % 16) == 0, "node tiles are exact");

typedef __attribute__((ext_vector_type(2))) float v2f;
typedef __attribute__((ext_vector_type(8))) float v8f;

// One async copy: every EXEC lane moves 4B  global -> LDS  (ASYNCcnt +1).
// NT hint keeps the streamed edge data from evicting L2-resident embeddings.
__device__ __forceinline__ void async_b32_to_lds(uint32_t lds_off, const void* gaddr) {
    asm volatile("global_load_async_to_lds_b32 %0, %1, off th:TH_LOAD_NT"
                 :: "v"(lds_off), "v"((uint64_t)(uintptr_t)gaddr)
                 : "memory");
}

__device__ __forceinline__ float atomic_add_f32_agent(float* p, float v) {
    return __hip_atomic_fetch_add(p, v, __ATOMIC_RELAXED, __HIP_MEMORY_SCOPE_AGENT);
}

// ---------------------------------------------------------------------------
// init: ego_a = concat(user, item); out = ego_a; ego_b = 0     (float4 lanes)
// ---------------------------------------------------------------------------
__global__ void lgcn_init(const float* __restrict__ user, const float* __restrict__ item,
                          float* __restrict__ ego, float* __restrict__ out,
                          float* __restrict__ nxt) {
    int i = blockIdx.x * blockDim.x + threadIdx.x;          // float4 index
    if (i < NELEM / 4) {
        const int u4 = USER_NUM * EMB_DIM / 4;              // 1,600,000
        float4 v = (i < u4) ? reinterpret_cast<const float4*>(user)[i]
                            : reinterpret_cast<const float4*>(item)[i - u4];
        reinterpret_cast<float4*>(ego)[i] = v;
        reinterpret_cast<float4*>(out)[i] = v;
        reinterpret_cast<float4*>(nxt)[i] = make_float4(0.f, 0.f, 0.f, 0.f);
    }
}

// ---------------------------------------------------------------------------
// SpMM: y[row] += val * x[col]  for each nnz.
// Per wave: double-buffered 32-edge batches staged in LDS via async loads.
// Per edge: 32 lanes read x[col] as float2 (256B coalesced, L2 hit),
//           2x global_atomic_add_f32 scatter (L2-resident RMW).
// ---------------------------------------------------------------------------
__global__ void lgcn_spmm(const int*   __restrict__ rows,
                          const int*   __restrict__ cols,
                          const float* __restrict__ vals,
                          const float* __restrict__ x,
                          float*       __restrict__ y) {
    __shared__ int   sh_row[WPB][2][BATCH];
    __shared__ int   sh_col[WPB][2][BATCH];
    __shared__ float sh_val[WPB][2][BATCH];

    const int lane   = threadIdx.x & 31;
    const int wave   = threadIdx.x >> 5;
    const int gwave  = blockIdx.x * WPB + wave;
    const int nwaves = gridDim.x * WPB;
    const int nbatch = NNZ / BATCH;                         // 150000

    // stage batch 'b' into buffer 'buf' (3 async ops -> ASYNCcnt +3)
    auto issue = [&](int buf, int b) {
        const int e = b * BATCH + lane;
        async_b32_to_lds((uint32_t)(uintptr_t)&sh_row[wave][buf][lane], rows + e);
        async_b32_to_lds((uint32_t)(uintptr_t)&sh_col[wave][buf][lane], cols + e);
        async_b32_to_lds((uint32_t)(uintptr_t)&sh_val[wave][buf][lane], vals + e);
    };

    int  b    = gwave;
    int  buf  = 0;
    bool have = (b < nbatch);
    if (have) issue(0, b);

    while (have) {
        const int  bn        = b + nwaves;                  // wave-uniform
        const bool have_next = (bn < nbatch);

        if (have_next) {
            issue(buf ^ 1, bn);
            // first 3 (current buffer) complete; next 3 still in flight
            asm volatile("s_wait_asynccnt 0x3" ::: "memory");
        } else {
            asm volatile("s_wait_asynccnt 0x0" ::: "memory");
        }

        #pragma unroll 4
        for (int j = 0; j < BATCH; ++j) {
            const int   r = sh_row[wave][buf][j];           // uniform broadcast
            const int   c = sh_col[wave][buf][j];
            const float v = sh_val[wave][buf][j];
            const float2 xv = *reinterpret_cast<const float2*>(
                x + (size_t)c * EMB_DIM + 2 * lane);
            float* yp = y + (size_t)r * EMB_DIM + 2 * lane;
            atomic_add_f32_agent(yp,     v * xv.x);
            atomic_add_f32_agent(yp + 1, v * xv.y);
        }

        buf ^= 1;
        b    = bn;
        have = have_next;
    }
}

// ---------------------------------------------------------------------------
// out += layer; tozero = 0   (between-layer bookkeeping, fused)
// ---------------------------------------------------------------------------
__global__ void lgcn_accum_zero(float* __restrict__ out, const float* __restrict__ layer,
                                float* __restrict__ tozero) {
    int i = blockIdx.x * blockDim.x + threadIdx.x;
    if (i < NELEM / 4) {
        float4 a = reinterpret_cast<float4*>(out)[i];
        float4 l = reinterpret_cast<const float4*>(layer)[i];
        a.x += l.x; a.y += l.y; a.z += l.z; a.w += l.w;
        reinterpret_cast<float4*>(out)[i] = a;
        reinterpret_cast<float4*>(tozero)[i] = make_float4(0.f, 0.f, 0.f, 0.f);
    }
}

// ---------------------------------------------------------------------------
// Final: out = 0.25 * (out + layer), one 16x16 tile per wave via chained
// v_wmma_f32_16x16x4_f32:   D = sum_k A_k x B_k,  A_k = 0.25 * I[:, 4k:4k+4].
// Layouts per ISA 7.12.2:
//   A 16x4 : lane -> M = lane%16 ; VGPR v -> K = v + 2*(lane/16)
//   B 4x16 : lane -> N = lane%16 ; VGPR v -> K = v + 2*(lane/16)
//   C/D    : lane -> N = lane%16 ; VGPR r -> M = r + 8*(lane/16)
// EXEC all-ones inside active waves (tile guard is wave-uniform).
// ---------------------------------------------------------------------------
__global__ void lgcn_final_wmma(float* __restrict__ out, const float* __restrict__ layer) {
    const int lane   = threadIdx.x & 31;
    const int wave   = threadIdx.x >> 5;
    const int gwave  = blockIdx.x * (blockDim.x >> 5) + wave;
    const int tilesN = EMB_DIM / 16;                        // 4
    const int tiles  = (N_NODES / 16) * tilesN;             // 37500
    if (gwave >= tiles) return;                             // wave-uniform exit

    const int m0     = (gwave / tilesN) * 16;
    const int n0     = (gwave % tilesN) * 16;
    const int half   = lane >> 4;                           // 0 | 1
    const int lanelo = lane & 15;

    v8f d = {};                                             // C starts at 0
    #pragma unroll
    for (int k = 0; k < 4; ++k) {
        v2f ak, bk;
        #pragma unroll
        for (int v = 0; v < 2; ++v) {
            const int K   = v + 2 * half;
            const int row = m0 + 4 * k + K;
            const size_t idx = (size_t)row * EMB_DIM + n0 + lanelo;
            bk[v] = out[idx] + layer[idx];                  // S = acc + layer3
            ak[v] = (lanelo == 4 * k + K) ? 0.25f : 0.0f;   // 0.25 * I slice
        }
        d = __builtin_amdgcn_wmma_f32_16x16x4_f32(
                /*neg_a=*/false, ak, /*neg_b=*/false, bk,
                /*c_mod=*/(short)0, d, /*reuse_a=*/false, /*reuse_b=*/false);
    }

    #pragma unroll
    for (int r = 0; r < 8; ++r)
        out[(size_t)(m0 + r + 8 * half) * EMB_DIM + n0 + lanelo] = d[r];
}

// ---------------------------------------------------------------------------
extern "C" void kernel_launch(void* const* d_in, const int* in_sizes, int n_in,
                              void* d_out, int out_size, void* d_ws, size_t ws_size,
                              hipStream_t stream) {
    const float* user = (const float*)d_in[0];
    const float* item = (const float*)d_in[1];
    const int*   row  = (const int*)  d_in[2];
    const int*   col  = (const int*)  d_in[3];
    const float* vals = (const float*)d_in[4];
    float*       out  = (float*)d_out;

    float* egoA = (float*)d_ws;            // 38.4 MB
    float* egoB = egoA + NELEM;            // 38.4 MB  (ws >= 76.8 MB)

    const int eblocks = (NELEM / 4 + 255) / 256;            // elementwise grids
    const int sblocks = 2048;                               // 16384 waves, ~9 batches each
    const int fblocks = (37500 + WPB - 1) / WPB;            // wmma tiles

    // acc = ego ; egoB = 0
    lgcn_init<<<eblocks, SPMM_BLOCK, 0, stream>>>(user, item, egoA, out, egoB);

    // layer 1: egoB = A*egoA ; acc += egoB ; egoA = 0
    lgcn_spmm<<<sblocks, SPMM_BLOCK, 0, stream>>>(row, col, vals, egoA, egoB);
    lgcn_accum_zero<<<eblocks, SPMM_BLOCK, 0, stream>>>(out, egoB, egoA);

    // layer 2: egoA = A*egoB ; acc += egoA ; egoB = 0
    lgcn_spmm<<<sblocks, SPMM_BLOCK, 0, stream>>>(row, col, vals, egoB, egoA);
    lgcn_accum_zero<<<eblocks, SPMM_BLOCK, 0, stream>>>(out, egoA, egoB);

    // layer 3: egoB = A*egoA ; out = 0.25*(acc + egoB)  [WMMA]
    lgcn_spmm<<<sblocks, SPMM_BLOCK, 0, stream>>>(row, col, vals, egoA, egoB);
    lgcn_final_wmma<<<fblocks, SPMM_BLOCK, 0, stream>>>(out, egoB);
}